// GNN_910533067627
// MI455X (gfx1250) — compile-verified
//
#include <hip/hip_runtime.h>

// CDNA5 / gfx1250: wave32, WMMA (not MFMA).
typedef __attribute__((ext_vector_type(2))) float v2f;
typedef __attribute__((ext_vector_type(8))) float v8f;

namespace {
constexpr int kN = 50000;   // nodes
constexpr int kE = 800000;  // edges
constexpr int kD = 64;      // input dim
constexpr int kH = 128;     // hidden dim
constexpr int kO = 64;      // output dim
}  // namespace

// ---------------------------------------------------------------------------
// Zero-fill (float4 vectorized, grid-stride).
// ---------------------------------------------------------------------------
__global__ void gnn_zero_kernel(float4* __restrict__ p, int n4) {
  int i = blockIdx.x * blockDim.x + threadIdx.x;
  int stride = gridDim.x * blockDim.x;
  for (; i < n4; i += stride) {
    p[i] = make_float4(0.f, 0.f, 0.f, 0.f);
  }
}

// ---------------------------------------------------------------------------
// Pack W[K x NOUT] row-major into k-pair-interleaved layout:
//   Wp[((k>>1)*NOUT + n)*2 + (k&1)] = W[k*NOUT + n]
// so a B fragment {W[ka][n], W[ka+1][n]} (ka even) is one aligned b64 load,
// coalesced across lanes.
// ---------------------------------------------------------------------------
__global__ void gnn_pack_w_kernel(const float* __restrict__ W,
                                  float* __restrict__ Wp, int K, int NOUT) {
  int t = blockIdx.x * blockDim.x + threadIdx.x;
  if (t >= K * NOUT) return;
  int k = t / NOUT;
  int n = t - k * NOUT;
  Wp[(((size_t)(k >> 1) * NOUT + n) << 1) + (k & 1)] = W[t];
}

// ---------------------------------------------------------------------------
// Edge scatter-add: agg[dst[e], :] += X[src[e], :]
// One thread handles one float4 chunk of one edge's feature vector.
// Destination tables are L2-resident -> global_atomic_add_f32 at L2 rate.
// ---------------------------------------------------------------------------
template <int D4>  // D4 = feature_dim / 4
__global__ void gnn_scatter_add_kernel(const float* __restrict__ X,
                                       const int* __restrict__ src,
                                       const int* __restrict__ dst,
                                       float* __restrict__ agg) {
  int t = blockIdx.x * blockDim.x + threadIdx.x;
  if (t >= kE * D4) return;
  int e = t / D4;            // power-of-two -> shift
  int c = t - e * D4;
  int s = src[e];
  int d = dst[e];
  float4 v = reinterpret_cast<const float4*>(X)[(size_t)s * D4 + c];
  float* p = agg + (((size_t)d * D4 + c) << 2);
  atomicAdd(p + 0, v.x);
  atomicAdd(p + 1, v.y);
  atomicAdd(p + 2, v.z);
  atomicAdd(p + 3, v.w);
}

// ---------------------------------------------------------------------------
// Fused dual GEMM + bias (+ ReLU):
//   Y = act(A1 @ W1 + A2 @ W2 + bias),  A: [N x K] row-major,
//   W1p/W2p: packed (K/2, NOUT, 2) layout from gnn_pack_w_kernel.
//
// One wave32 computes a 16x32 tile of Y (two adjacent 16x16 N-tiles) with
// V_WMMA_F32_16X16X4_F32: the A fragment is loaded once per k-step and reused
// by both WMMAs; each B fragment is a single coalesced b64 load.
//
// VGPR layouts per CDNA5 ISA (7.12.2):
//   A 16x4 f32 : lane m = lane&15, v2f = {A[m][k0+2*half], A[m][k0+2*half+1]}
//   B 4x16 f32 : lane n = lane&15, v2f = {B[k0+2*half][n], B[k0+2*half+1][n]}
//   C/D 16x16  : lane n = lane&15, c[r] = C[r + 8*half][n]
// ---------------------------------------------------------------------------
template <int K, int NOUT, bool RELU>
__global__ void gnn_gemm2x_wmma_kernel(const float* __restrict__ A1,
                                       const float* __restrict__ W1p,
                                       const float* __restrict__ A2,
                                       const float* __restrict__ W2p,
                                       const float* __restrict__ bias,
                                       float* __restrict__ Y,
                                       int num_tiles2) {
  constexpr int TILES_N2 = NOUT / 32;  // 16x32 double-tiles per row
  const int wave_in_block = threadIdx.x >> 5;                 // wave32
  const int tile = blockIdx.x * (blockDim.x >> 5) + wave_in_block;
  if (tile >= num_tiles2) return;  // wave-uniform: EXEC stays all-1s for WMMA

  const int lane = threadIdx.x & 31;
  const int lm = lane & 15;
  const int half = lane >> 4;

  const int tm = tile / TILES_N2;
  const int tn = tile - tm * TILES_N2;
  const int row  = tm * 16 + lm;        // A-matrix row this lane owns
  const int col0 = tn * 32 + lm;        // B/C/D column, left 16x16 tile
  const int col1 = col0 + 16;           // right 16x16 tile

  // Packed weights are tiny (<= 64 KB total) and shared by every wave.
  __builtin_prefetch(W1p + (((size_t)half * NOUT + col0) << 1), 0, 0);
  __builtin_prefetch(W2p + (((size_t)half * NOUT + col0) << 1), 0, 0);

  v8f c0 = {};
  v8f c1 = {};

#pragma unroll
  for (int kk = 0; kk < K; kk += 4) {
    const int ka = kk + half * 2;       // even
    const int p  = ka >> 1;             // k-pair index
    v2f a  = *reinterpret_cast<const v2f*>(A1 + (size_t)row * K + ka);
    v2f b0 = *reinterpret_cast<const v2f*>(W1p + (((size_t)p * NOUT + col0) << 1));
    v2f b1 = *reinterpret_cast<const v2f*>(W1p + (((size_t)p * NOUT + col1) << 1));
    c0 = __builtin_amdgcn_wmma_f32_16x16x4_f32(false, a, false, b0,
                                               (short)0, c0, false, false);
    c1 = __builtin_amdgcn_wmma_f32_16x16x4_f32(false, a, false, b1,
                                               (short)0, c1, false, false);
  }
#pragma unroll
  for (int kk = 0; kk < K; kk += 4) {
    const int ka = kk + half * 2;
    const int p  = ka >> 1;
    v2f a  = *reinterpret_cast<const v2f*>(A2 + (size_t)row * K + ka);
    v2f b0 = *reinterpret_cast<const v2f*>(W2p + (((size_t)p * NOUT + col0) << 1));
    v2f b1 = *reinterpret_cast<const v2f*>(W2p + (((size_t)p * NOUT + col1) << 1));
    c0 = __builtin_amdgcn_wmma_f32_16x16x4_f32(false, a, false, b0,
                                               (short)0, c0, false, false);
    c1 = __builtin_amdgcn_wmma_f32_16x16x4_f32(false, a, false, b1,
                                               (short)0, c1, false, false);
  }

  const float bv0 = bias[col0];
  const float bv1 = bias[col1];
#pragma unroll
  for (int r = 0; r < 8; ++r) {
    const size_t m = (size_t)(tm * 16 + half * 8 + r);
    float v0 = c0[r] + bv0;
    float v1 = c1[r] + bv1;
    if (RELU) {
      v0 = fmaxf(v0, 0.0f);
      v1 = fmaxf(v1, 0.0f);
    }
    Y[m * NOUT + col0] = v0;
    Y[m * NOUT + col1] = v1;
  }
}

// ---------------------------------------------------------------------------
// Launch: zero+pack -> scatter1 -> gemm1(+relu) -> scatter2 -> gemm2 -> d_out
// ---------------------------------------------------------------------------
extern "C" void kernel_launch(void* const* d_in, const int* in_sizes, int n_in,
                              void* d_out, int out_size, void* d_ws, size_t ws_size,
                              hipStream_t stream) {
  (void)in_sizes; (void)n_in; (void)out_size; (void)ws_size;

  const float* x       = (const float*)d_in[0];
  const int*   ei      = (const int*)d_in[1];   // [2, E] row-major
  const float* W1_rel  = (const float*)d_in[2];
  const float* W1_root = (const float*)d_in[3];
  const float* b1      = (const float*)d_in[4];
  const float* W2_rel  = (const float*)d_in[5];
  const float* W2_root = (const float*)d_in[6];
  const float* b2      = (const float*)d_in[7];
  float* out = (float*)d_out;

  const int* src = ei;        // edge_index[0]
  const int* dst = ei + kE;   // edge_index[1]

  // Workspace layout (fp32):
  //   agg1[N*D] | h[N*H] | agg2[N*H] | w1rp[D*H] | w1tp[D*H] | w2rp[H*O] | w2tp[H*O]
  float* agg1 = (float*)d_ws;
  float* h    = agg1 + (size_t)kN * kD;
  float* agg2 = h    + (size_t)kN * kH;
  float* w1rp = agg2 + (size_t)kN * kH;   // packed W1_rel
  float* w1tp = w1rp + (size_t)kD * kH;   // packed W1_root
  float* w2rp = w1tp + (size_t)kD * kH;   // packed W2_rel
  float* w2tp = w2rp + (size_t)kH * kO;   // packed W2_root

  // Zero the accumulation buffers (h is fully overwritten by gemm1).
  gnn_zero_kernel<<<2048, 256, 0, stream>>>((float4*)agg1, kN * kD / 4);
  gnn_zero_kernel<<<2048, 256, 0, stream>>>((float4*)agg2, kN * kH / 4);

  // Pack the four weight matrices for b64 B-fragment loads.
  gnn_pack_w_kernel<<<(kD * kH + 255) / 256, 256, 0, stream>>>(W1_rel,  w1rp, kD, kH);
  gnn_pack_w_kernel<<<(kD * kH + 255) / 256, 256, 0, stream>>>(W1_root, w1tp, kD, kH);
  gnn_pack_w_kernel<<<(kH * kO + 255) / 256, 256, 0, stream>>>(W2_rel,  w2rp, kH, kO);
  gnn_pack_w_kernel<<<(kH * kO + 255) / 256, 256, 0, stream>>>(W2_root, w2tp, kH, kO);

  // Layer 1 scatter: agg1[dst] += x[src]
  {
    const int total = kE * (kD / 4);
    gnn_scatter_add_kernel<kD / 4>
        <<<(total + 255) / 256, 256, 0, stream>>>(x, src, dst, agg1);
  }
  // Layer 1 fused GEMM: h = relu(agg1 @ W1_rel + x @ W1_root + b1)
  {
    const int tiles2 = (kN / 16) * (kH / 32);  // 12500
    gnn_gemm2x_wmma_kernel<kD, kH, true>
        <<<(tiles2 + 7) / 8, 256, 0, stream>>>(agg1, w1rp, x, w1tp, b1, h, tiles2);
  }
  // Layer 2 scatter: agg2[dst] += h[src]
  {
    const int total = kE * (kH / 4);
    gnn_scatter_add_kernel<kH / 4>
        <<<(total + 255) / 256, 256, 0, stream>>>(h, src, dst, agg2);
  }
  // Layer 2 fused GEMM: out = agg2 @ W2_rel + h @ W2_root + b2
  {
    const int tiles2 = (kN / 16) * (kO / 32);  // 6250
    gnn_gemm2x_wmma_kernel<kH, kO, false>
        <<<(tiles2 + 7) / 8, 256, 0, stream>>>(agg2, w2rp, h, w2tp, b2, out, tiles2);
  }
}